// PointNetWithFPS_81965155876949
// MI455X (gfx1250) — compile-verified
//
#include <hip/hip_runtime.h>
#include <hip/hip_bf16.h>

// ---------------------------------------------------------------------------
// PointNet-with-FPS for MI455X (gfx1250, wave32).
//
// Phase 1 (fps_kernel): dominant cost, serialization-bound. One 1024-thread
//   block per batch; 16 points/thread held in VGPRs (x,y,z,dist = 64 regs),
//   per-step argmax via wave32 shfl_xor reduction + one 32-entry LDS stage
//   (2 barriers per step). Winning coords ride through the reduction so the
//   next centroid needs no gather.
// Phase 2 (gemm_kernel): gather + [65536x68]x[68x128] fp32 GEMM with
//   v_wmma_f32_16x16x4_f32 (exact fp32). W staged in LDS in a K-pair-
//   interleaved layout so every B fragment is ONE aligned ds_load_b64
//   (no register repacking). Bias pre-loaded into accumulators, ReLU fused.
// ---------------------------------------------------------------------------

typedef __attribute__((ext_vector_type(2))) float v2f;
typedef __attribute__((ext_vector_type(8))) float v8f;

#define B_    32
#define N_    16384
#define S_    2048
#define F_    64
#define DOUT_ 128
#define K_    68          // 64 feats + 3 xyz, padded to 68 (multiple of 4)
#define PTS_PER_THR 16
#define THREADS1 1024

__global__ __launch_bounds__(1024) void fps_kernel(
    const float* __restrict__ xyz, const int* __restrict__ init_farthest,
    float* __restrict__ new_xyz, int* __restrict__ idx_out)
{
  const int b   = blockIdx.x;
  const int tid = threadIdx.x;
  const float* bxyz = xyz + (size_t)b * N_ * 3;

  float px[PTS_PER_THR], py[PTS_PER_THR], pz[PTS_PER_THR], dist[PTS_PER_THR];
#pragma unroll
  for (int j = 0; j < PTS_PER_THR; ++j) {
    int p = tid + j * THREADS1;                 // strided -> coalesced b96 loads
    px[j] = bxyz[p * 3 + 0];
    py[j] = bxyz[p * 3 + 1];
    pz[j] = bxyz[p * 3 + 2];
    dist[j] = 1e10f;
  }

  __shared__ float redD[32], redX[32], redY[32], redZ[32];
  __shared__ int   redI[32];
  __shared__ float cX, cY, cZ;
  __shared__ int   cI;

  // seed centroid from init_farthest (owner thread publishes its coords)
  int f0 = init_farthest[b];
  if ((f0 & (THREADS1 - 1)) == tid) {
    int j = f0 >> 10;
    cX = px[j]; cY = py[j]; cZ = pz[j]; cI = f0;
  }
  __syncthreads();

  const int wave = tid >> 5;
  const int lane = tid & 31;

  for (int s = 0; s < S_; ++s) {
    const float cx = cX, cy = cY, cz = cZ;
    const int   ci = cI;
    if (tid == 0) {                              // record BEFORE update (torch loop)
      size_t o = (size_t)(b * S_ + s);
      new_xyz[o * 3 + 0] = cx;
      new_xyz[o * 3 + 1] = cy;
      new_xyz[o * 3 + 2] = cz;
      idx_out[o] = ci;
    }

    // distance update + thread-local argmax (first-index tie-break)
    float bd = -1.0f; int bi = 0x7fffffff;
    float bxv = 0.f, byv = 0.f, bzv = 0.f;
#pragma unroll
    for (int j = 0; j < PTS_PER_THR; ++j) {
      float dx = px[j] - cx, dy = py[j] - cy, dz = pz[j] - cz;
      float d  = dx * dx + dy * dy + dz * dz;
      float nd = fminf(dist[j], d);
      dist[j]  = nd;
      int gi = tid + j * THREADS1;
      if (nd > bd || (nd == bd && gi < bi)) { bd = nd; bi = gi; bxv = px[j]; byv = py[j]; bzv = pz[j]; }
    }

    // wave32 shuffle reduction carrying (dist, idx, x, y, z)
#pragma unroll
    for (int off = 16; off > 0; off >>= 1) {
      float od = __shfl_xor(bd,  off, 32);
      int   oi = __shfl_xor(bi,  off, 32);
      float ox = __shfl_xor(bxv, off, 32);
      float oy = __shfl_xor(byv, off, 32);
      float oz = __shfl_xor(bzv, off, 32);
      if (od > bd || (od == bd && oi < bi)) { bd = od; bi = oi; bxv = ox; byv = oy; bzv = oz; }
    }
    if (lane == 0) { redD[wave] = bd; redI[wave] = bi; redX[wave] = bxv; redY[wave] = byv; redZ[wave] = bzv; }
    __syncthreads();

    if (tid < 32) {                              // wave 0 reduces the 32 wave winners
      bd = redD[tid]; bi = redI[tid]; bxv = redX[tid]; byv = redY[tid]; bzv = redZ[tid];
#pragma unroll
      for (int off = 16; off > 0; off >>= 1) {
        float od = __shfl_xor(bd,  off, 32);
        int   oi = __shfl_xor(bi,  off, 32);
        float ox = __shfl_xor(bxv, off, 32);
        float oy = __shfl_xor(byv, off, 32);
        float oz = __shfl_xor(bzv, off, 32);
        if (od > bd || (od == bd && oi < bi)) { bd = od; bi = oi; bxv = ox; byv = oy; bzv = oz; }
      }
      if (tid == 0) { cX = bxv; cY = byv; cZ = bzv; cI = bi; }
    }
    __syncthreads();
  }
}

// ---------------------------------------------------------------------------
// Gather + GEMM + bias + ReLU via V_WMMA_F32_16X16X4_F32.
// One wave = one 16-row M-tile x all 8 N-tiles of 16 (A fragment reused x8).
// W LDS layout is K-pair interleaved: Wp[(k>>1)*256 + col*2 + (k&1)], so a
// B fragment {W[k][col], W[k+1][col]} is a single aligned ds_load_b64.
// ---------------------------------------------------------------------------
#define WAVES2   8
#define THREADS2 256

__global__ __launch_bounds__(256) void gemm_kernel(
    const float* __restrict__ xyz,  const float* __restrict__ feats,
    const float* __restrict__ W,    const float* __restrict__ bias,
    const int*   __restrict__ idx,  float* __restrict__ out)
{
  __shared__ float Wp[K_ * DOUT_];               // 68 x 128 = 34 KB, pair-interleaved
  const int tid = threadIdx.x;
  for (int i = tid; i < K_ * DOUT_; i += THREADS2) {
    int k   = i >> 7;                            // source row (K)
    int col = i & 127;
    float v = (k < K_ - 1) ? W[i] : 0.0f;        // pad row 67 with zeros
    Wp[(k >> 1) * (2 * DOUT_) + col * 2 + (k & 1)] = v;
  }
  __syncthreads();

  const int wave = tid >> 5;
  const int lane = tid & 31;
  const int half = lane >> 4;                    // 0: K pair {4t,4t+1}, 1: {4t+2,4t+3}
  const int l16  = lane & 15;

  const int m_base = (blockIdx.x * WAVES2 + wave) * 16;

  // per-lane gathered row (A-matrix M = l16 for both lane halves)
  const int row   = m_base + l16;
  const int b     = row >> 11;                   // /S_
  const int myidx = idx[row];
  const float* fp = feats + ((size_t)b * N_ + myidx) * F_;
  const float* xp = xyz   + ((size_t)b * N_ + myidx) * 3;

  v8f acc[8];
#pragma unroll
  for (int n = 0; n < 8; ++n) {                  // seed accumulators with bias
    float bv = bias[n * 16 + l16];
    v8f c = {bv, bv, bv, bv, bv, bv, bv, bv};
    acc[n] = c;
  }

  // 16 full feature chunks (K = 0..63)
  for (int t = 0; t < 16; ++t) {
    const int k = t * 4 + half * 2;              // even
    v2f a = *(const v2f*)(fp + k);               // global_load_b64 gather
    const float* bp = Wp + (k >> 1) * (2 * DOUT_);
#pragma unroll
    for (int n = 0; n < 8; ++n) {
      v2f bf = *(const v2f*)(bp + (n * 16 + l16) * 2);   // one ds_load_b64
      acc[n] = __builtin_amdgcn_wmma_f32_16x16x4_f32(
          false, a, false, bf, (short)0, acc[n], false, false);
    }
  }

  // last chunk: K = 64..67 -> xyz (3) + zero pad
  {
    const int k = 64 + half * 2;
    v2f a;
    if (half == 0) { a.x = xp[0]; a.y = xp[1]; }
    else           { a.x = xp[2]; a.y = 0.0f;  }
    const float* bp = Wp + (k >> 1) * (2 * DOUT_);
#pragma unroll
    for (int n = 0; n < 8; ++n) {
      v2f bf = *(const v2f*)(bp + (n * 16 + l16) * 2);
      acc[n] = __builtin_amdgcn_wmma_f32_16x16x4_f32(
          false, a, false, bf, (short)0, acc[n], false, false);
    }
  }

  // ReLU + store. C/D layout: VGPR r -> M = r (lanes 0-15) / r+8 (lanes 16-31)
#pragma unroll
  for (int n = 0; n < 8; ++n) {
    const int col = n * 16 + l16;
#pragma unroll
    for (int r = 0; r < 8; ++r) {
      float v = fmaxf(acc[n][r], 0.0f);
      int orow = m_base + r + 8 * half;
      out[(size_t)orow * DOUT_ + col] = v;
    }
  }
}

extern "C" void kernel_launch(void* const* d_in, const int* in_sizes, int n_in,
                              void* d_out, int out_size, void* d_ws, size_t ws_size,
                              hipStream_t stream) {
  const float* xyz   = (const float*)d_in[0];
  const float* feats = (const float*)d_in[1];
  const int*   initf = (const int*)d_in[2];
  const float* W     = (const float*)d_in[3];
  const float* bias  = (const float*)d_in[4];

  float* out_all   = (float*)d_out;
  float* new_xyz   = out_all;                       // [32,2048,3]
  float* conv_out  = out_all + (size_t)B_ * S_ * 3; // [32,2048,128]
  int*   idxws     = (int*)d_ws;                    // [32*2048] indices

  fps_kernel<<<B_, THREADS1, 0, stream>>>(xyz, initf, new_xyz, idxws);

  const int mtiles = (B_ * S_) / 16;                // 4096 M-tiles
  gemm_kernel<<<mtiles / WAVES2, THREADS2, 0, stream>>>(
      xyz, feats, W, bias, idxws, conv_out);
}